// SpatialCrossAttention_18794776887980
// MI455X (gfx1250) — compile-verified
//
#include <hip/hip_runtime.h>
#include <hip/hip_bf16.h>

typedef float v2f __attribute__((ext_vector_type(2)));
typedef float v8f __attribute__((ext_vector_type(8)));

#define NHEADS 4
#define NPOINTS 8
#define BB 2
#define SS 6
#define NN 10000
#define CC 128
#define HH 32
#define WW 88
#define MM (HH * WW)      // 2816
#define NROWS (BB * NN)   // 20000

// LDS weight tile: k-pair interleaved, padded so (stride mod 64) == 32
// lds[(k>>1)*STR + c*2 + (k&1)]  ->  B-fragment = one aligned ds_load_b64
#define STR_A 288   // 128 cols: 2*128 + 32
#define STR_B 224   // 96 cols:  2*96  + 32

// ---------------- Stage A: value projection (M x C) @ (C x C), WMMA f32 16x16x4 ----------------
// vproj layout: [bs = b*S+s][head][m][hd=32]  -> each (head,m) row is a contiguous 128B gather target
__global__ __launch_bounds__(256)
void value_proj_kernel(const float* __restrict__ value,
                       const float* __restrict__ Wv,
                       const float* __restrict__ bv,
                       float* __restrict__ vproj) {
    __shared__ float ldsB[64 * STR_A];
    const int tid = threadIdx.x;
    for (int idx = tid; idx < 128 * 128; idx += 256) {
        int k = idx >> 7, c = idx & 127;
        ldsB[(k >> 1) * STR_A + c * 2 + (k & 1)] = Wv[idx];
    }
    __syncthreads();
    const int wave = tid >> 5;
    const int lane = tid & 31;
    const int half = lane >> 4;
    const int lx   = lane & 15;
    const int bs = blockIdx.y;          // b*S + s
    const int b = bs / SS, s = bs % SS;
    const int mbase = blockIdx.x * 128 + wave * 16;
    // value layout (S, M, B, C)
    const float* arow = value + (((size_t)s * MM + (size_t)(mbase + lx)) * BB + b) * CC;
    v8f acc[8];
#pragma unroll
    for (int ct = 0; ct < 8; ++ct) acc[ct] = (v8f){0.f,0.f,0.f,0.f,0.f,0.f,0.f,0.f};
    for (int k0 = 0; k0 < 128; k0 += 4) {
        v2f a = *(const v2f*)&arow[k0 + 2 * half];
        const float* brow = &ldsB[(k0 / 2 + half) * STR_A + lx * 2];
#pragma unroll
        for (int ct = 0; ct < 8; ++ct) {
            v2f bf = *(const v2f*)&brow[ct * 32];
            acc[ct] = __builtin_amdgcn_wmma_f32_16x16x4_f32(
                false, a, false, bf, (short)0, acc[ct], false, false);
        }
    }
#pragma unroll
    for (int ct = 0; ct < 8; ++ct) {
        int c = ct * 16 + lx;
        int head = c >> 5;
        int hd = c & 31;
        float bias = bv[c];
        float* dst = vproj + (((size_t)bs * NHEADS + head) * MM) * 32 + hd;
#pragma unroll
        for (int r = 0; r < 8; ++r) {
            int m = mbase + r + 8 * half;
            dst[(size_t)m * 32] = acc[ct][r] + bias;
        }
    }
}

// ---------------- Stage B: (q+qpos) @ [Woff | Wa]  (20000x128 @ 128x96) ----------------
__global__ __launch_bounds__(256)
void qproj_kernel(const float* __restrict__ query, const float* __restrict__ query_pos,
                  const float* __restrict__ Woff, const float* __restrict__ boff,
                  const float* __restrict__ Wa, const float* __restrict__ ba,
                  float* __restrict__ offs, float* __restrict__ attn) {
    __shared__ float ldsB[64 * STR_B];
    const int tid = threadIdx.x;
    for (int idx = tid; idx < 128 * 96; idx += 256) {
        int k = idx / 96, c = idx % 96;
        ldsB[(k >> 1) * STR_B + c * 2 + (k & 1)] =
            (c < 64) ? Woff[k * 64 + c] : Wa[k * 32 + (c - 64)];
    }
    __syncthreads();
    const int wave = tid >> 5, lane = tid & 31, half = lane >> 4, lx = lane & 15;
    const int rowbase = blockIdx.x * 128 + wave * 16;
    if (rowbase >= NROWS) return;    // 20000 % 16 == 0, whole-wave guard
    const float* q  = query + (size_t)(rowbase + lx) * CC;
    const float* qp = query_pos + (size_t)(rowbase + lx) * CC;
    v8f acc[6];
#pragma unroll
    for (int ct = 0; ct < 6; ++ct) acc[ct] = (v8f){0.f,0.f,0.f,0.f,0.f,0.f,0.f,0.f};
    for (int k0 = 0; k0 < 128; k0 += 4) {
        v2f a0 = *(const v2f*)&q[k0 + 2 * half];
        v2f a1 = *(const v2f*)&qp[k0 + 2 * half];
        v2f a = a0 + a1;
        const float* brow = &ldsB[(k0 / 2 + half) * STR_B + lx * 2];
#pragma unroll
        for (int ct = 0; ct < 6; ++ct) {
            v2f bf = *(const v2f*)&brow[ct * 32];
            acc[ct] = __builtin_amdgcn_wmma_f32_16x16x4_f32(
                false, a, false, bf, (short)0, acc[ct], false, false);
        }
    }
#pragma unroll
    for (int ct = 0; ct < 6; ++ct) {
        int c = ct * 16 + lx;
#pragma unroll
        for (int r = 0; r < 8; ++r) {
            int row = rowbase + r + 8 * half;
            float v = acc[ct][r];
            if (c < 64) offs[(size_t)row * 64 + c] = v + boff[c];
            else        attn[(size_t)row * 32 + (c - 64)] = v + ba[c - 64];
        }
    }
}

// ---------------- Stage C: deformable sampling + camera reduction ----------------
// One wave per (b,n). lane = hd channel; lane j also stores (off,aw) pair j=h*8+p, broadcast via shfl.
__global__ __launch_bounds__(256)
void sample_kernel(const float* __restrict__ vproj, const float* __restrict__ offs,
                   const float* __restrict__ attn, const float* __restrict__ refpts,
                   const unsigned int* __restrict__ bev, float* __restrict__ slots) {
    const int lane = threadIdx.x & 31;
    const int wid = blockIdx.x * 8 + (threadIdx.x >> 5);
    if (wid >= NROWS) return;
    const int b = wid / NN, n = wid % NN;

    const float2* offp = (const float2*)offs;
    float2 myoff = offp[(size_t)wid * 32 + lane];
    float logit = attn[(size_t)wid * 32 + lane];
    // softmax over 8 points within each head (groups of 8 lanes)
    float mx = logit;
    mx = fmaxf(mx, __shfl_xor(mx, 1, 8));
    mx = fmaxf(mx, __shfl_xor(mx, 2, 8));
    mx = fmaxf(mx, __shfl_xor(mx, 4, 8));
    float e = __expf(logit - mx);
    float sum = e;
    sum += __shfl_xor(sum, 1, 8);
    sum += __shfl_xor(sum, 2, 8);
    sum += __shfl_xor(sum, 4, 8);
    float aw = e / sum;

    // count: per (b,n) number of cameras with any hit (per-batch mask)
    int cnt = 0;
#pragma unroll
    for (int s = 0; s < SS; ++s)
        cnt += (bev[((size_t)s * BB + b) * NN + n] != 0u) ? 1 : 0;
    float inv = 1.0f / (float)(cnt < 1 ? 1 : cnt);

    float acc[NHEADS] = {0.f, 0.f, 0.f, 0.f};
    for (int s = 0; s < SS; ++s) {
        // sel uses batch-0 mask (as in reference); skip camera if sel==0
        if (bev[((size_t)s * BB + 0) * NN + n] == 0u) continue;
        const float2* rp = (const float2*)refpts;  // (S,B,N,D,2)
        float2 r4[4];
#pragma unroll
        for (int d = 0; d < 4; ++d)
            r4[d] = rp[(((size_t)s * BB + b) * NN + n) * 4 + d];
        const size_t vbase = ((size_t)(b * SS + s) * NHEADS) * MM * 32;
#pragma unroll
        for (int h = 0; h < NHEADS; ++h) {
            const float* vh = vproj + vbase + (size_t)h * MM * 32 + lane;
#pragma unroll
            for (int p = 0; p < NPOINTS; ++p) {
                int j = h * 8 + p;
                float ox = __shfl(myoff.x, j, 32);
                float oy = __shfl(myoff.y, j, 32);
                float w  = __shfl(aw, j, 32);
                float2 rr = r4[p & 3];
                float x = rr.x * (float)WW + ox - 0.5f;
                float y = rr.y * (float)HH + oy - 0.5f;
                float x0f = floorf(x), y0f = floorf(y);
                float fx = x - x0f, fy = y - y0f;
                int x0 = (int)x0f, y0 = (int)y0f;
                float sampv = 0.f;
#pragma unroll
                for (int corner = 0; corner < 4; ++corner) {
                    int dy = corner >> 1, dx = corner & 1;
                    int yy = y0 + dy, xx = x0 + dx;
                    float wy = dy ? fy : (1.f - fy);
                    float wx = dx ? fx : (1.f - fx);
                    bool valid = (yy >= 0) & (yy < HH) & (xx >= 0) & (xx < WW);
                    int yc = min(max(yy, 0), HH - 1);
                    int xc = min(max(xx, 0), WW - 1);
                    float g = vh[(size_t)(yc * WW + xc) * 32];   // 128B coalesced row
                    sampv += (valid ? wy * wx : 0.f) * g;
                }
                acc[h] += sampv * w;
            }
        }
    }
    float* outp = slots + (size_t)wid * CC + lane;
#pragma unroll
    for (int h = 0; h < NHEADS; ++h) outp[h * 32] = acc[h] * inv;
}

// ---------------- Stage D: output projection + bias + residual ----------------
__global__ __launch_bounds__(256)
void out_proj_kernel(const float* __restrict__ slots, const float* __restrict__ Wout,
                     const float* __restrict__ bout, const float* __restrict__ query,
                     float* __restrict__ out) {
    __shared__ float ldsB[64 * STR_A];
    const int tid = threadIdx.x;
    for (int idx = tid; idx < 128 * 128; idx += 256) {
        int k = idx >> 7, c = idx & 127;
        ldsB[(k >> 1) * STR_A + c * 2 + (k & 1)] = Wout[idx];
    }
    __syncthreads();
    const int wave = tid >> 5, lane = tid & 31, half = lane >> 4, lx = lane & 15;
    const int rowbase = blockIdx.x * 128 + wave * 16;
    if (rowbase >= NROWS) return;
    const float* arow = slots + (size_t)(rowbase + lx) * CC;
    v8f acc[8];
#pragma unroll
    for (int ct = 0; ct < 8; ++ct) acc[ct] = (v8f){0.f,0.f,0.f,0.f,0.f,0.f,0.f,0.f};
    for (int k0 = 0; k0 < 128; k0 += 4) {
        v2f a = *(const v2f*)&arow[k0 + 2 * half];
        const float* brow = &ldsB[(k0 / 2 + half) * STR_A + lx * 2];
#pragma unroll
        for (int ct = 0; ct < 8; ++ct) {
            v2f bf = *(const v2f*)&brow[ct * 32];
            acc[ct] = __builtin_amdgcn_wmma_f32_16x16x4_f32(
                false, a, false, bf, (short)0, acc[ct], false, false);
        }
    }
#pragma unroll
    for (int ct = 0; ct < 8; ++ct) {
        int c = ct * 16 + lx;
        float bias = bout[c];
#pragma unroll
        for (int r = 0; r < 8; ++r) {
            int row = rowbase + r + 8 * half;
            out[(size_t)row * CC + c] = acc[ct][r] + bias + query[(size_t)row * CC + c];
        }
    }
}

extern "C" void kernel_launch(void* const* d_in, const int* in_sizes, int n_in,
                              void* d_out, int out_size, void* d_ws, size_t ws_size,
                              hipStream_t stream) {
    const float* query     = (const float*)d_in[0];
    // d_in[1] = key (unused by the reference math)
    const float* value     = (const float*)d_in[2];
    const float* query_pos = (const float*)d_in[3];
    const float* refpts    = (const float*)d_in[4];
    const unsigned int* bev = (const unsigned int*)d_in[5];  // 4 bool bytes per (s,b,n)
    // d_in[6] = spatial_shapes (static: H=32, W=88, hardcoded)
    const float* Wv   = (const float*)d_in[7];
    const float* bv   = (const float*)d_in[8];
    const float* Woff = (const float*)d_in[9];
    const float* boff = (const float*)d_in[10];
    const float* Wa   = (const float*)d_in[11];
    const float* ba   = (const float*)d_in[12];
    const float* Wout = (const float*)d_in[13];
    const float* bout = (const float*)d_in[14];
    float* out = (float*)d_out;

    // workspace carve-up (floats): vproj 17.3MB | offs 5.1MB | attn 2.6MB | slots 10.2MB
    float* vproj = (float*)d_ws;
    float* offs  = vproj + (size_t)BB * SS * NHEADS * MM * 32;
    float* attn  = offs + (size_t)NROWS * 64;
    float* slots = attn + (size_t)NROWS * 32;

    value_proj_kernel<<<dim3(MM / 128, BB * SS), 256, 0, stream>>>(value, Wv, bv, vproj);
    qproj_kernel<<<dim3((NROWS + 127) / 128), 256, 0, stream>>>(query, query_pos, Woff, boff,
                                                                 Wa, ba, offs, attn);
    sample_kernel<<<dim3(NROWS / 8), 256, 0, stream>>>(vproj, offs, attn, refpts, bev, slots);
    out_proj_kernel<<<dim3((NROWS + 127) / 128), 256, 0, stream>>>(slots, Wout, bout, query, out);
}